// Head_37718402794047
// MI455X (gfx1250) — compile-verified
//
#include <hip/hip_runtime.h>
#include <hip/hip_bf16.h>
#include <stdint.h>

#define SCORE_THRESH 0.25f
#define NMS_THRESH   0.45f
#define DETECTIONS   100
#define K_CAND       300
#define KPAD         304          // 19 * 16, WMMA-tile padded
#define MAXC         1024         // candidate cap per image (bitonic-sorted)
#define NLOC         25200        // 80*80*3 + 40*40*3 + 20*20*3
#define NCLS         80
#define NIMG         16

typedef __attribute__((ext_vector_type(2))) float v2f;
typedef __attribute__((ext_vector_type(8))) float v8f;

__device__ __forceinline__ float sigmoidf_(float x) {
  return 1.0f / (1.0f + __expf(-x));
}

// CDNA5 async global->LDS copy (ASYNCcnt path, CDNA5 ISA)
__device__ __forceinline__ void async_b128_to_lds(uint32_t lds_off, const void* g) {
  asm volatile("global_load_async_to_lds_b128 %0, %1, off"
               :: "v"(lds_off), "v"(g) : "memory");
}
__device__ __forceinline__ void wait_asynccnt0() {
  asm volatile("s_wait_asynccnt 0" ::: "memory");
}

// ---------------------------------------------------------------------------
// Kernel 0: reset per-image candidate counters (ws is not re-poisoned by the
// harness, so this must run every call).
// ---------------------------------------------------------------------------
__global__ void zero_kernel(int* cand_cnt) {
  if (threadIdx.x < NIMG) cand_cnt[threadIdx.x] = 0;
}

// ---------------------------------------------------------------------------
// Kernel A: fused sigmoid + box decode + score threshold + candidate compaction.
// One thread per (image, location). Streams 137 MB once (HBM-roofline bound,
// ~6 us at 23.3 TB/s); writes boxes (6.4 MB) and only thresholded
// (score, flat-index) candidates via atomic compaction — avoids materializing
// the 129 MB [B, N*C] score tensor implied by the reference's top_k.
// ---------------------------------------------------------------------------
__global__ void decode_kernel(const float* __restrict__ p3,
                              const float* __restrict__ p4,
                              const float* __restrict__ p5,
                              const float* __restrict__ anchors,
                              float* __restrict__ boxes_all,
                              float* __restrict__ cand_score,
                              int*   __restrict__ cand_idx,
                              int*   __restrict__ cand_cnt) {
  int gid = blockIdx.x * blockDim.x + threadIdx.x;
  if (gid >= NIMG * NLOC) return;
  int b = gid / NLOC;
  int n = gid - b * NLOC;

  const float* p; float stride; int W; int m; int lvl;
  if (n < 19200)      { lvl = 0; m = n;         W = 80; stride =  8.f; p = p3 + (size_t)b * 19200 * 85; }
  else if (n < 24000) { lvl = 1; m = n - 19200; W = 40; stride = 16.f; p = p4 + (size_t)b *  4800 * 85; }
  else                { lvl = 2; m = n - 24000; W = 20; stride = 32.f; p = p5 + (size_t)b *  1200 * 85; }

  int a    = m % 3;
  int cell = m / 3;
  int x    = cell % W;
  int y    = cell / W;
  const float* q = p + (size_t)m * 85;
  __builtin_prefetch(q + 85 * 64, 0, 1);   // -> global_prefetch_b8

  float sx  = sigmoidf_(q[0]);
  float sy  = sigmoidf_(q[1]);
  float sw  = sigmoidf_(q[2]);
  float sh  = sigmoidf_(q[3]);
  float obj = sigmoidf_(q[4]);

  float aw = anchors[(lvl * 3 + a) * 2 + 0];
  float ah = anchors[(lvl * 3 + a) * 2 + 1];
  float cx = (2.f * sx - 0.5f + (float)x) * stride;
  float cy = (2.f * sy - 0.5f + (float)y) * stride;
  float ww = 4.f * sw * sw * aw;
  float hh = 4.f * sh * sh * ah;

  float4 box = make_float4(cx - 0.5f * ww, cy - 0.5f * hh,
                           cx + 0.5f * ww, cy + 0.5f * hh);
  ((float4*)boxes_all)[(size_t)b * NLOC + n] = box;

  if (obj > SCORE_THRESH) {
    // s = obj*sigmoid(l) > T  <=>  sigmoid(l) > T/obj; keep direct form (1 exp each)
    #pragma unroll 4
    for (int c = 0; c < NCLS; ++c) {
      float s = obj * sigmoidf_(q[5 + c]);
      if (s > SCORE_THRESH) {
        int pos = atomicAdd(&cand_cnt[b], 1);
        if (pos < MAXC) {
          cand_score[b * MAXC + pos] = s;
          cand_idx  [b * MAXC + pos] = n * NCLS + c;
        }
      }
    }
  }
}

// ---------------------------------------------------------------------------
// Kernel B: per-image top-K via LDS bitonic sort of compacted candidates,
// then gather + clip boxes and build class-offset NMS boxes.
// One workgroup (512 threads = 16 waves) per image.
// ---------------------------------------------------------------------------
__global__ void topk_kernel(const float* __restrict__ boxes_all,
                            const float* __restrict__ cand_score,
                            const int*   __restrict__ cand_idx,
                            const int*   __restrict__ cand_cnt,
                            const int*   __restrict__ image_shapes,
                            const int*   __restrict__ max_size_p,
                            float* __restrict__ cbox,
                            float* __restrict__ nbox,
                            float* __restrict__ cscore,
                            int*   __restrict__ clabel) {
  int img = blockIdx.x;
  __shared__ float sk[MAXC];
  __shared__ int   sv[MAXC];

  int cnt = cand_cnt[img];
  if (cnt > MAXC) cnt = MAXC;

  for (int i = threadIdx.x; i < MAXC; i += blockDim.x) {
    if (i < cnt) { sk[i] = cand_score[img * MAXC + i]; sv[i] = cand_idx[img * MAXC + i]; }
    else         { sk[i] = -1.0f;                       sv[i] = 0; }
  }

  // bitonic sort, descending
  for (unsigned k = 2; k <= MAXC; k <<= 1) {
    for (unsigned j = k >> 1; j > 0; j >>= 1) {
      __syncthreads();
      for (unsigned i = threadIdx.x; i < MAXC; i += blockDim.x) {
        unsigned ixj = i ^ j;
        if (ixj > i) {
          bool desc = ((i & k) == 0);
          bool sw = desc ? (sk[i] < sk[ixj]) : (sk[i] > sk[ixj]);
          if (sw) {
            float tk = sk[i]; sk[i] = sk[ixj]; sk[ixj] = tk;
            int   tv = sv[i]; sv[i] = sv[ixj]; sv[ixj] = tv;
          }
        }
      }
    }
  }
  __syncthreads();

  float w   = (float)image_shapes[img * 2 + 1];
  float h   = (float)image_shapes[img * 2 + 0];
  float msz = (float)max_size_p[0];

  for (int t = threadIdx.x; t < KPAD; t += blockDim.x) {
    float  sc = 0.f;
    float4 cb = make_float4(0.f, 0.f, 0.f, 0.f);
    int    lab = 0;
    if (t < K_CAND && t < cnt && sk[t] > 0.f) {
      sc = sk[t];
      int idx  = sv[t];
      int aidx = idx / NCLS;
      lab      = idx - aidx * NCLS;
      const float* bb = boxes_all + ((size_t)img * NLOC + aidx) * 4;
      cb.x = fminf(fmaxf(bb[0], 0.f), w);
      cb.y = fminf(fmaxf(bb[1], 0.f), h);
      cb.z = fminf(fmaxf(bb[2], 0.f), w);
      cb.w = fminf(fmaxf(bb[3], 0.f), h);
    }
    size_t o = (size_t)img * KPAD + t;
    ((float4*)cbox)[o] = cb;
    float off = (float)lab * msz;                 // batched-NMS class offset
    ((float4*)nbox)[o] = make_float4(cb.x + off, cb.y + off, cb.z + off, cb.w + off);
    cscore[o] = sc;
    clabel[o] = lab;
  }
}

// ---------------------------------------------------------------------------
// Kernel C: per-image NMS. Async-copies class-offset boxes into LDS, builds
// the 304x304 suppression bitmask in 16x16 tiles where the area[i]+area[j]
// outer-sum per tile is a rank-2 f32 matmul done with V_WMMA_F32_16X16X4_F32
// (A = [area_i, 1], B = [1; area_j]); min/max intersection is VALU.
// The IoU threshold is evaluated division-free:
//   inter/(asum - inter + eps) > T  <=>  inter*(1+T) > T*(asum + eps)
// (valid since asum - inter >= 0 here), removing the 11-instruction IEEE
// divide sequence from the inner loop. Then greedy keep + top-100 emit.
// ---------------------------------------------------------------------------
__global__ void nms_kernel(const float* __restrict__ cbox,
                           const float* __restrict__ nbox,
                           const float* __restrict__ cscore,
                           const int*   __restrict__ clabel,
                           const float* __restrict__ scale_factors,
                           float* __restrict__ out) {
  int img = blockIdx.x;
  __shared__ float    s_nb[KPAD * 4];
  __shared__ float    s_area[KPAD];
  __shared__ float    s_score[KPAD];
  __shared__ unsigned s_supp[KPAD * 10];   // 304 rows x 320 bits

  int tid = threadIdx.x;

  // async global -> LDS stage of the NMS boxes (ASYNCcnt), overlapped with
  // the regular score loads and suppression-matrix zeroing below.
  const float* gnb = nbox + (size_t)img * KPAD * 4;
  for (int i = tid; i < KPAD; i += blockDim.x) {
    uint32_t lds_off = (uint32_t)(uintptr_t)(&s_nb[i * 4]);
    async_b128_to_lds(lds_off, gnb + i * 4);
  }
  for (int i = tid; i < KPAD; i += blockDim.x)
    s_score[i] = cscore[(size_t)img * KPAD + i];
  for (int i = tid; i < KPAD * 10; i += blockDim.x)
    s_supp[i] = 0u;
  wait_asynccnt0();
  __syncthreads();

  for (int i = tid; i < KPAD; i += blockDim.x) {
    float bw = fmaxf(s_nb[i * 4 + 2] - s_nb[i * 4 + 0], 0.f);
    float bh = fmaxf(s_nb[i * 4 + 3] - s_nb[i * 4 + 1], 0.f);
    s_area[i] = bw * bh;
  }
  __syncthreads();

  const int TI = KPAD / 16;                 // 19 tiles per side
  int      wave  = tid >> 5;
  unsigned lane  = tid & 31u;
  int      nwave = blockDim.x >> 5;

  for (int t = wave; t < TI * TI; t += nwave) {
    int ti = t / TI, tj = t - ti * TI;
    int i0 = ti * 16, j0 = tj * 16;

    // rank-2 fragments: D[m][n] = area[i0+m]*1 + 1*area[j0+n]
    float ai = s_area[i0 + (lane & 15)];
    float aj = s_area[j0 + (lane & 15)];
    bool  lo = lane < 16;                   // lanes >=16 carry K=2,3 -> zero
    v2f afrag; afrag.x = lo ? ai : 0.f;  afrag.y = lo ? 1.f : 0.f;
    v2f bfrag; bfrag.x = lo ? 1.f : 0.f; bfrag.y = lo ? aj  : 0.f;
    v8f acc = {};
    acc = __builtin_amdgcn_wmma_f32_16x16x4_f32(false, afrag, false, bfrag,
                                                (short)0, acc, false, false);

    int j     = j0 + (int)(lane & 15);
    int mbase = i0 + (int)((lane >> 4) << 3);
    float jx1 = s_nb[j * 4 + 0], jy1 = s_nb[j * 4 + 1];
    float jx2 = s_nb[j * 4 + 2], jy2 = s_nb[j * 4 + 3];

    #pragma unroll
    for (int r = 0; r < 8; ++r) {
      int i = mbase + r;
      if (j > i) {
        float lx = fmaxf(s_nb[i * 4 + 0], jx1);
        float ly = fmaxf(s_nb[i * 4 + 1], jy1);
        float rx = fminf(s_nb[i * 4 + 2], jx2);
        float ry = fminf(s_nb[i * 4 + 3], jy2);
        float iw = fmaxf(rx - lx, 0.f);
        float ih = fmaxf(ry - ly, 0.f);
        float inter = iw * ih;
        // division-free: inter/(asum - inter + eps) > T
        if (inter * (1.0f + NMS_THRESH) > NMS_THRESH * (acc[r] + 1e-9f))
          atomicOr(&s_supp[i * 10 + (j >> 5)], 1u << (j & 31));
      }
    }
  }
  __syncthreads();

  // greedy keep + emit (tiny serial tail: 300 iters x 10 words)
  if (tid == 0) {
    unsigned keep[10];
    #pragma unroll
    for (int w = 0; w < 10; ++w) keep[w] = 0xFFFFFFFFu;
    keep[9] = (1u << (K_CAND - 288)) - 1u;   // only bits 288..299 valid
    for (int i = 0; i < K_CAND; ++i) {
      bool act = ((keep[i >> 5] >> (i & 31)) & 1u) && (s_score[i] > SCORE_THRESH);
      if (act) {
        #pragma unroll
        for (int w = 0; w < 10; ++w) keep[w] &= ~s_supp[i * 10 + w];
      }
    }
    float inv = 1.0f / scale_factors[img];
    float* o  = out + (size_t)img * DETECTIONS * 6;
    int cnt = 0;
    for (int i = 0; i < K_CAND && cnt < DETECTIONS; ++i) {
      if (((keep[i >> 5] >> (i & 31)) & 1u) && (s_score[i] > SCORE_THRESH)) {
        const float* cb = cbox + ((size_t)img * KPAD + i) * 4;
        o[cnt * 6 + 0] = cb[0] * inv;
        o[cnt * 6 + 1] = cb[1] * inv;
        o[cnt * 6 + 2] = cb[2] * inv;
        o[cnt * 6 + 3] = cb[3] * inv;
        o[cnt * 6 + 4] = s_score[i];
        o[cnt * 6 + 5] = (float)clabel[(size_t)img * KPAD + i];
        ++cnt;
      }
    }
    for (; cnt < DETECTIONS; ++cnt)
      for (int k = 0; k < 6; ++k) o[cnt * 6 + k] = 0.f;
  }
}

// ---------------------------------------------------------------------------
extern "C" void kernel_launch(void* const* d_in, const int* in_sizes, int n_in,
                              void* d_out, int out_size, void* d_ws, size_t ws_size,
                              hipStream_t stream) {
  const float* p3            = (const float*)d_in[0];
  const float* p4            = (const float*)d_in[1];
  const float* p5            = (const float*)d_in[2];
  const float* anchors       = (const float*)d_in[3];
  const int*   image_shapes  = (const int*)  d_in[4];
  const float* scale_factors = (const float*)d_in[5];
  const int*   max_size_p    = (const int*)  d_in[6];
  float* out = (float*)d_out;

  // workspace layout (all offsets 256B-aligned)
  char* ws = (char*)d_ws;
  size_t off = 0;
  int*   cand_cnt   = (int*)  (ws + off); off += 256;
  float* boxes_all  = (float*)(ws + off); off += (size_t)NIMG * NLOC * 4 * sizeof(float); // 6,451,200 B
  float* cand_score = (float*)(ws + off); off += (size_t)NIMG * MAXC * sizeof(float);
  int*   cand_idx   = (int*)  (ws + off); off += (size_t)NIMG * MAXC * sizeof(int);
  float* cbox       = (float*)(ws + off); off += (size_t)NIMG * KPAD * 4 * sizeof(float);
  float* nbox       = (float*)(ws + off); off += (size_t)NIMG * KPAD * 4 * sizeof(float);
  float* cscore     = (float*)(ws + off); off += (size_t)NIMG * KPAD * sizeof(float);
  int*   clabel     = (int*)  (ws + off); off += (size_t)NIMG * KPAD * sizeof(int);

  zero_kernel<<<1, 32, 0, stream>>>(cand_cnt);

  int total = NIMG * NLOC;
  decode_kernel<<<(total + 255) / 256, 256, 0, stream>>>(
      p3, p4, p5, anchors, boxes_all, cand_score, cand_idx, cand_cnt);

  topk_kernel<<<NIMG, 512, 0, stream>>>(
      boxes_all, cand_score, cand_idx, cand_cnt, image_shapes, max_size_p,
      cbox, nbox, cscore, clabel);

  nms_kernel<<<NIMG, 256, 0, stream>>>(
      cbox, nbox, cscore, clabel, scale_factors, out);

  (void)in_sizes; (void)n_in; (void)out_size; (void)ws_size;
}